// MixtureCogrammar_91113436217844
// MI455X (gfx1250) — compile-verified
//
#include <hip/hip_runtime.h>
#include <hip/hip_bf16.h>

// Problem dims (fixed by the reference)
#define B_DIM   1024
#define D_DIM   256
#define N_DIM   256
#define DM_DIM  128
#define V_DIM   512
#define COLS    (D_DIM * N_DIM)   // 65536 flattened (d,n) columns
#define NCOL    64                // columns per workgroup panel
#define PSTRIDE 160               // floats per K-pair row: 128 data + 32 pad
                                  // (PSTRIDE % 64 == 32 -> the two lane halves
                                  //  hit disjoint LDS bank halves)

typedef float v2f __attribute__((ext_vector_type(2)));
typedef float v8f __attribute__((ext_vector_type(8)));

// ---------------------------------------------------------------------------
// Kernel 1: attn = softmax(morphosyn @ W_affix, axis=-1)   [B, V]
// One block per batch row; thread t owns columns t and t+256.
// ---------------------------------------------------------------------------
__global__ __launch_bounds__(256)
void attn_kernel(const float* __restrict__ morphosyn,
                 const float* __restrict__ W,
                 float* __restrict__ attn)
{
    __shared__ float ms[DM_DIM];
    __shared__ float red[256];
    const int b = blockIdx.x, tid = threadIdx.x;

    if (tid < DM_DIM) ms[tid] = morphosyn[b * DM_DIM + tid];
    __syncthreads();

    float s0 = 0.f, s1 = 0.f;
#pragma unroll 8
    for (int k = 0; k < DM_DIM; ++k) {
        const float m = ms[k];
        s0 = fmaf(m, W[k * V_DIM + tid], s0);
        s1 = fmaf(m, W[k * V_DIM + tid + 256], s1);
    }

    // block max
    red[tid] = fmaxf(s0, s1); __syncthreads();
    for (int s = 128; s > 0; s >>= 1) {
        if (tid < s) red[tid] = fmaxf(red[tid], red[tid + s]);
        __syncthreads();
    }
    const float mx = red[0]; __syncthreads();

    const float e0 = __expf(s0 - mx), e1 = __expf(s1 - mx);
    red[tid] = e0 + e1; __syncthreads();
    for (int s = 128; s > 0; s >>= 1) {
        if (tid < s) red[tid] += red[tid + s];
        __syncthreads();
    }
    const float inv = 1.f / red[0];

    attn[b * V_DIM + tid]       = e0 * inv;
    attn[b * V_DIM + tid + 256] = e1 * inv;
}

// ---------------------------------------------------------------------------
// Kernel 2: wC[b,n] = sum_{i,j,k} a_i b_j f_k * cumsum(softmax(pivot[i,j,b,k,:]))[n]
// One block per batch; thread n = position. Block softmax + Hillis-Steele scan.
// ---------------------------------------------------------------------------
__global__ __launch_bounds__(256)
void wc_kernel(const float* __restrict__ logits,
               const float* __restrict__ alpha,
               const float* __restrict__ beta,
               const float* __restrict__ phi,
               float* __restrict__ wC)
{
    __shared__ float red[256];
    const int b = blockIdx.x, n = threadIdx.x;

    // tiny softmaxes, computed redundantly per thread
    float wa[2], wb[2], wf[5];
    {
        float a0 = alpha[0], a1 = alpha[1];
        float m = fmaxf(a0, a1);
        float e0 = __expf(a0 - m), e1 = __expf(a1 - m);
        float s = e0 + e1; wa[0] = e0 / s; wa[1] = e1 / s;
    }
    {
        float b0 = beta[0], b1 = beta[1];
        float m = fmaxf(b0, b1);
        float e0 = __expf(b0 - m), e1 = __expf(b1 - m);
        float s = e0 + e1; wb[0] = e0 / s; wb[1] = e1 / s;
    }
    {
        float p[5], m = -1e30f, s = 0.f;
        for (int k = 0; k < 5; ++k) { p[k] = phi[k]; m = fmaxf(m, p[k]); }
        for (int k = 0; k < 5; ++k) { p[k] = __expf(p[k] - m); s += p[k]; }
        for (int k = 0; k < 5; ++k) wf[k] = p[k] / s;
    }

    float acc = 0.f;
    for (int i = 0; i < 2; ++i)
    for (int j = 0; j < 2; ++j)
    for (int k = 0; k < 5; ++k) {
        const size_t off =
            ((((size_t)(i * 2 + j) * B_DIM + b) * 5 + k) * N_DIM) + n;
        const float x = logits[off];

        // block max
        red[n] = x; __syncthreads();
        for (int s = 128; s > 0; s >>= 1) {
            if (n < s) red[n] = fmaxf(red[n], red[n + s]);
            __syncthreads();
        }
        const float mx = red[0]; __syncthreads();
        const float e = __expf(x - mx);

        // block sum
        red[n] = e; __syncthreads();
        for (int s = 128; s > 0; s >>= 1) {
            if (n < s) red[n] += red[n + s];
            __syncthreads();
        }
        const float S = red[0]; __syncthreads();

        // inclusive scan of e
        red[n] = e; __syncthreads();
        for (int o = 1; o < 256; o <<= 1) {
            const float t = (n >= o) ? red[n - o] : 0.f;
            __syncthreads();
            red[n] += t;
            __syncthreads();
        }
        const float C = red[n] / S;
        acc = fmaf(wa[i] * wb[j] * wf[k], C, acc);
        __syncthreads();   // protect red[] before next combo
    }

    wC[b * N_DIM + n] = acc;
}

// ---------------------------------------------------------------------------
// Kernel 3: fused  affix = attn @ affix_vocab  (fp32 WMMA 16x16x4)
//           out   = stem + wC * (affix - stem)
//
// Grid: 1024 blocks, each owns a 64-column panel (all 1024 rows).
// The affix_vocab panel (K rows 0..255 x 64 cols fp32) is staged in LDS once per
// block in K-PAIR-INTERLEAVED form:
//     element (k, c) lives at panel[(k>>1)*PSTRIDE + 2*c + (k&1)]
// so every WMMA B-fragment {(kb,c), (kb+1,c)} is ONE aligned ds_load_b64
// straight into an even VGPR pair -> no repack moves, half the DS traffic.
// PSTRIDE % 64 == 32 puts the two lane halves (pair-rows p and p+1) on
// disjoint LDS bank halves -> conflict-free b64 reads.
// ---------------------------------------------------------------------------
__global__ __launch_bounds__(256)
void gemm_blend_kernel(const float* __restrict__ attn,
                       const float* __restrict__ vocab,
                       const float* __restrict__ stem,
                       const float* __restrict__ wC,
                       float* __restrict__ out)
{
    __shared__ float panel[(V_DIM / 2) * PSTRIDE];   // 256*160*4 = 160 KB

    const int tid = threadIdx.x;
    const int c0  = blockIdx.x * NCOL;

    // Stage panel: task i -> K-pair p = i>>5, column-pair c2 = i&31.
    // Load float2 from rows 2p and 2p+1, store interleaved float4:
    //   {v0.x, v1.x, v0.y, v1.y} at panel[p*PSTRIDE + 4*c2]
    for (int i = tid; i < (V_DIM / 2) * (NCOL / 2) / 2; i += 256) {
        const int p  = i >> 5;          // 0..127
        const int c2 = i & 31;          // 0..31 (pairs of columns)
        const float2 v0 = *reinterpret_cast<const float2*>(
            vocab + (size_t)(2 * p)     * COLS + c0 + 2 * c2);
        const float2 v1 = *reinterpret_cast<const float2*>(
            vocab + (size_t)(2 * p + 1) * COLS + c0 + 2 * c2);
        float4 w;
        w.x = v0.x; w.y = v1.x; w.z = v0.y; w.w = v1.y;
        *reinterpret_cast<float4*>(&panel[p * PSTRIDE + 4 * c2]) = w;
    }
    __syncthreads();

    const int lane = tid & 31;
    const int wave = tid >> 5;
    const int half = lane >> 4;    // 0: lanes 0-15, 1: lanes 16-31
    const int l16  = lane & 15;

    for (int rb = 0; rb < 8; ++rb) {
        const int row0 = (wave * 8 + rb) * 16;       // 16-row block of batch dim
        v8f acc0 = {}; v8f acc1 = {}; v8f acc2 = {}; v8f acc3 = {};

        // A fragment source: attn row (row0 + l16); lane half picks K {0,1}/{2,3}
        const float* aptr = attn + (size_t)(row0 + l16) * V_DIM + 2 * half;

        for (int k0 = 0; k0 < V_DIM; k0 += 4) {
            // A: contiguous pair -> one global_load_b64
            const v2f a = *reinterpret_cast<const v2f*>(aptr + k0);

            // B: one ds_load_b64 per column tile (pair-row = k0/2 + half)
            const float* pb = &panel[(k0 / 2 + half) * PSTRIDE + 2 * l16];
            const v2f b0 = *reinterpret_cast<const v2f*>(pb);
            const v2f b1 = *reinterpret_cast<const v2f*>(pb + 32);
            const v2f b2 = *reinterpret_cast<const v2f*>(pb + 64);
            const v2f b3 = *reinterpret_cast<const v2f*>(pb + 96);

            acc0 = __builtin_amdgcn_wmma_f32_16x16x4_f32(
                false, a, false, b0, (short)0, acc0, false, false);
            acc1 = __builtin_amdgcn_wmma_f32_16x16x4_f32(
                false, a, false, b1, (short)0, acc1, false, false);
            acc2 = __builtin_amdgcn_wmma_f32_16x16x4_f32(
                false, a, false, b2, (short)0, acc2, false, false);
            acc3 = __builtin_amdgcn_wmma_f32_16x16x4_f32(
                false, a, false, b3, (short)0, acc3, false, false);
        }

        // Fused epilogue: C/D layout -> row = row0 + g + 8*half, col = c0+16t+l16
#pragma unroll
        for (int g = 0; g < 8; ++g) {
            const int    row   = row0 + g + 8 * half;
            const size_t rbase = (size_t)row * COLS;
            const int    wrow  = row * N_DIM;
#pragma unroll
            for (int t = 0; t < 4; ++t) {
                const int   col = c0 + 16 * t + l16;
                const float w   = wC[wrow + (col & (N_DIM - 1))];
                const float s   = stem[rbase + col];
                const float af  = (t == 0) ? acc0[g]
                                : (t == 1) ? acc1[g]
                                : (t == 2) ? acc2[g]
                                :            acc3[g];
                out[rbase + col] = fmaf(w, af - s, s);   // stem + w*(affix-stem)
            }
        }
    }
}

// ---------------------------------------------------------------------------
// Launch. Workspace layout: attn (2 MB fp32) | wC (1 MB fp32).
// ---------------------------------------------------------------------------
extern "C" void kernel_launch(void* const* d_in, const int* in_sizes, int n_in,
                              void* d_out, int out_size, void* d_ws, size_t ws_size,
                              hipStream_t stream)
{
    const float* stem   = (const float*)d_in[0];   // [B, D, N]
    const float* morpho = (const float*)d_in[1];   // [B, DM]
    const float* plog   = (const float*)d_in[2];   // [2, 2, B, 5, N]
    const float* Waff   = (const float*)d_in[3];   // [DM, V]
    const float* vocab  = (const float*)d_in[4];   // [V, D, N]
    const float* alpha  = (const float*)d_in[5];   // [2]
    const float* beta   = (const float*)d_in[6];   // [2]
    const float* phi    = (const float*)d_in[7];   // [5]
    float* outp = (float*)d_out;

    float* attn = (float*)d_ws;                            // B*V = 524288 floats
    float* wc   = attn + (size_t)B_DIM * V_DIM;            // B*N = 262144 floats

    attn_kernel<<<B_DIM, 256, 0, stream>>>(morpho, Waff, attn);
    wc_kernel<<<B_DIM, 256, 0, stream>>>(plog, alpha, beta, phi, wc);
    gemm_blend_kernel<<<COLS / NCOL, 256, 0, stream>>>(attn, vocab, stem, wc, outp);
}